// Sinkhorn_64441689309638
// MI455X (gfx1250) — compile-verified
//
#include <hip/hip_runtime.h>

// Sinkhorn (B=128, N=513) for MI455X / gfx1250.
//
// Roofline: K = s+eps is 134 MB total -> resident in 192 MB L2. 100 iters x 2
// streaming GEMV passes = ~27 GB L2 traffic (the bound); compute is only ~27
// GFLOP f32. Every K element is single-use per pass, so the iteration kernel
// is pure coalesced streaming (no LDS staging / no tensor-core benefit there).
// The epilogue T = u (x) v .* K uses one v_wmma_f32_16x16x4_f32 rank-1 outer
// product per 16x16 tile per wave.

#define NN       513
#define NM1      512
#define EPSK     1e-4f
#define STOPTHR  1e-3f
#define MAXITER  100

typedef __attribute__((ext_vector_type(2))) float v2f;
typedef __attribute__((ext_vector_type(8))) float v8f;

// ---------------------------------------------------------------------------
// Kernel 1: 100 Sinkhorn iterations. One block per batch (freeze/early-stop is
// batch-uniform -> uniform break). 512 threads = 16 waves.
// ---------------------------------------------------------------------------
__global__ __launch_bounds__(512) void sinkhorn_iter(const float* __restrict__ s,
                                                     float* __restrict__ u_out,
                                                     float* __restrict__ v_out)
{
    __shared__ float su[NN];     // u vector
    __shared__ float sv[NN];     // v vector
    __shared__ float dvp[512];   // per-thread |dv| partials
    __shared__ float dup[16];    // per-wave |du| partials
    __shared__ float serr;

    const int tid  = threadIdx.x;
    const int lane = tid & 31;
    const int wave = tid >> 5;
    const int b    = blockIdx.x;
    const float* __restrict__ Kb = s + (size_t)b * NN * NN;

    for (int i = tid; i < NN; i += 512) { su[i] = 1.f; sv[i] = 1.f; }
    __syncthreads();

    for (int it = 0; it < MAXITER; ++it) {
        // ---- Pass A: t[m] = sum_n K[n,m]*u[n];  v_new[m] = b[m]/t[m]
        // Thread j owns column m=j (block reads full contiguous rows: coalesced).
        float dv_local = 0.f;
        for (int m = tid; m < NN; m += 512) {
            const float* col = Kb + m;
            float t = 0.f;
            for (int n0 = 0; n0 < 512; n0 += 8) {
                __builtin_prefetch(col + (size_t)(n0 + 32) * NN, 0, 1);  // global_prefetch_b8
                #pragma unroll
                for (int k = 0; k < 8; ++k)
                    t = fmaf(col[(size_t)(n0 + k) * NN] + EPSK, su[n0 + k], t);
            }
            t = fmaf(col[(size_t)512 * NN] + EPSK, su[512], t);
            const float bm = (m == NM1) ? (float)NM1 : 1.f;     // dustbin marginal
            const float vn = bm / t;
            dv_local += fabsf(sv[m] - vn);
            sv[m] = vn;                                         // commit v := v_new
        }
        dvp[tid] = dv_local;
        __syncthreads();

        // ---- Pass B: p[n] = sum_m K[n,m]*v[m];  u_new[n] = a[n]/p[n]
        // One wave per row: 32 contiguous lanes along the row (coalesced),
        // shfl-xor tree reduction.
        float du_local = 0.f;
        for (int n = wave; n < NN; n += 16) {
            const float* row = Kb + (size_t)n * NN;
            __builtin_prefetch(row + 16 * NN + lane * 16, 0, 1);  // cover next row of this wave
            float p = 0.f;
            for (int m = lane; m < NN; m += 32)
                p = fmaf(row[m] + EPSK, sv[m], p);
            #pragma unroll
            for (int off = 16; off > 0; off >>= 1)
                p += __shfl_xor(p, off, 32);
            if (lane == 0) {
                const float an = (n == NM1) ? (float)NM1 : 1.f;
                const float un = an / p;
                du_local += fabsf(su[n] - un);
                su[n] = un;                                     // commit u := u_new
            }
        }
        if (lane == 0) dup[wave] = du_local;
        __syncthreads();

        // ---- Deterministic error reduction: err = (sum|du|)^2 + (sum|dv|)^2
        if (wave == 0) {
            float dv = 0.f;
            #pragma unroll
            for (int k = 0; k < 16; ++k) dv += dvp[lane + 32 * k];
            #pragma unroll
            for (int off = 16; off > 0; off >>= 1)
                dv += __shfl_xor(dv, off, 32);
            if (lane == 0) {
                float du = 0.f;
                #pragma unroll
                for (int k = 0; k < 16; ++k) du += dup[k];
                serr = du * du + dv * dv;
            }
        }
        __syncthreads();
        if (serr <= STOPTHR) break;   // uniform: emulates per-batch freezing
    }

    for (int i = tid; i < NN; i += 512) {
        u_out[(size_t)b * NN + i] = su[i];
        v_out[(size_t)b * NN + i] = sv[i];
    }
}

// ---------------------------------------------------------------------------
// Kernel 2: T[b,n,m] = u[n]*(s[n,m]+eps)*v[m] for n,m < 512.
// One wave per 16x16 tile; rank-1 outer product via v_wmma_f32_16x16x4_f32:
//   A = u-tile in column k=0 (16x4), B = v-tile in row k=0 (4x16), C = 0
//   => D[i][j] = u[n0+i]*v[m0+j]
// D layout: VGPR r, lane L -> row r + 8*(L>=16), col L&15 -> 64B-contiguous
// half-wave loads/stores for K and T.
// ---------------------------------------------------------------------------
__global__ __launch_bounds__(256) void outer_wmma(const float* __restrict__ s,
                                                  const float* __restrict__ u_all,
                                                  const float* __restrict__ v_all,
                                                  float* __restrict__ out)
{
    const int lane = threadIdx.x & 31;
    const int gw   = blockIdx.x * 8 + (threadIdx.x >> 5);  // global wave = tile id
    const int bidx = gw >> 10;                             // 32*32 = 1024 tiles/batch
    const int t    = gw & 1023;
    const int n0   = (t >> 5) << 4;
    const int m0   = (t & 31) << 4;

    const float* u = u_all + (size_t)bidx * NN;
    const float* v = v_all + (size_t)bidx * NN;

    // A 16x4 f32: v0 lanes0-15 = A[M][0] (=u), lanes16-31 = A[M][2] (=0); v1 = 0
    v2f A;  A.x  = (lane < 16) ? u[n0 + lane] : 0.f;  A.y  = 0.f;
    // B 4x16 f32: v0 lanes0-15 = B[0][j] (=v), other rows zero
    v2f Bm; Bm.x = (lane < 16) ? v[m0 + lane] : 0.f;  Bm.y = 0.f;
    v8f C = {0.f, 0.f, 0.f, 0.f, 0.f, 0.f, 0.f, 0.f};

    v8f D = __builtin_amdgcn_wmma_f32_16x16x4_f32(
        /*neg_a=*/false, A, /*neg_b=*/false, Bm,
        /*c_mod=*/(short)0, C, /*reuse_a=*/false, /*reuse_b=*/false);

    const float* Kb = s   + (size_t)bidx * NN * NN;
    float*       Ob = out + (size_t)bidx * NM1 * NM1;
    const int col   = m0 + (lane & 15);
    const int rbase = n0 + ((lane >> 4) << 3);
    #pragma unroll
    for (int r = 0; r < 8; ++r) {
        const int row = rbase + r;
        const float kv = Kb[(size_t)row * NN + col] + EPSK;
        Ob[(size_t)row * NM1 + col] = D[r] * kv;
    }
}

extern "C" void kernel_launch(void* const* d_in, const int* in_sizes, int n_in,
                              void* d_out, int out_size, void* d_ws, size_t ws_size,
                              hipStream_t stream)
{
    (void)n_in; (void)out_size; (void)ws_size;
    const float* s = (const float*)d_in[0];
    const int B = in_sizes[1];                 // nrows has one entry per batch

    float* u_ws = (float*)d_ws;                // B*513 floats
    float* v_ws = u_ws + (size_t)B * NN;       // B*513 floats

    sinkhorn_iter<<<B, 512, 0, stream>>>(s, u_ws, v_ws);

    const int tiles  = B * 32 * 32;            // 16x16 tiles covering 512x512
    const int blocks = tiles / 8;              // 8 waves (tiles) per 256-thread block
    outer_wmma<<<blocks, 256, 0, stream>>>(s, u_ws, v_ws, (float*)d_out);
}